// EideticLinearLayer_22239340659213
// MI455X (gfx1250) — compile-verified
//
#include <hip/hip_runtime.h>
#include <math.h>

// ---------------------------------------------------------------------------
// EideticLinearLayer for MI455X (gfx1250, wave32, WMMA)
//   out     = x @ W^T + bias                         [16384, 4096] f32
//   indices = searchsorted(quantiles[col], wx, left) [16384, 4096] f32
// d_out = [out ; indices] concatenated flat.
//
// - V_WMMA_F32_16X16X4_F32 (exact f32 -> integer-exact searchsorted indices)
// - 128x128 block tile, 8 waves, 64x32 per wave (4x2 WMMA accumulators)
// - Double-buffered LDS K-chunks (1 barrier per chunk)
// - Non-temporal output stores: 536 MB write stream must not evict the
//   L2-resident x (16.8 MB, reused 32x) / W (4 MB, reused 128x) / quantiles.
// ---------------------------------------------------------------------------

typedef __attribute__((ext_vector_type(2))) float v2f;
typedef __attribute__((ext_vector_type(8))) float v8f;

#define B_DIM 16384
#define K_DIM 256
#define N_DIM 4096
#define QB    31

#define BM 128
#define BN 128
#define BK 16
#define NCH (K_DIM / BK)  // 16 K-chunks
#define LDT (BK + 4)      // 20 floats = 80 B pitch: 16B-aligned, conflict-free
#define QP  33            // quantile row pitch (31 bounds + INF sentinel + pad)

__launch_bounds__(256)
__global__ void eidetic_linear_kernel(const float* __restrict__ x,
                                      const float* __restrict__ w,
                                      const float* __restrict__ bias,
                                      const float* __restrict__ quant,
                                      float* __restrict__ out)
{
    __shared__ float As[2][BM * LDT];
    __shared__ float Bs[2][BN * LDT];
    __shared__ float Qs[BN * QP];

    const int tid  = threadIdx.x;        // 0..255 (8 waves)
    const int wid  = tid >> 5;
    const int lane = tid & 31;
    const int half = lane >> 4;          // 0: lanes 0-15, 1: lanes 16-31
    const int l16  = lane & 15;

    const int wm = wid >> 2;             // 0..1  -> 64-row slab
    const int wn = wid & 3;              // 0..3  -> 32-col slab

    const int n0 = blockIdx.x * BN;
    const int m0 = blockIdx.y * BM;

    // loader mapping: 512 float4 per 128x16 tile, 2 per thread
    const int row0 = tid >> 2;                    // rows 0..63
    const int row1 = (tid + 256) >> 2;            // rows 64..127
    const int c4   = (tid & 3) << 2;              // 0,4,8,12

    // ---- stage per-column quantile boundaries into LDS (+INF sentinel) ----
    for (int i = tid; i < BN * 32; i += 256) {
        const int c = i >> 5, j = i & 31;
        Qs[c * QP + j] = (j < QB) ? quant[(size_t)(n0 + c) * QB + j]
                                  : __builtin_inff();
    }

    // ---- accumulators: 4 (M) x 2 (N) tiles of 16x16 f32 ----
    v8f acc[4][2];
    const v8f zacc = {};
#pragma unroll
    for (int mt = 0; mt < 4; ++mt)
#pragma unroll
        for (int nt = 0; nt < 2; ++nt)
            acc[mt][nt] = zacc;

    // ---- preload K-chunk 0 into buffer 0 ----
    {
        const float4 a0 = *(const float4*)&x[(size_t)(m0 + row0) * K_DIM + c4];
        const float4 a1 = *(const float4*)&x[(size_t)(m0 + row1) * K_DIM + c4];
        const float4 b0 = *(const float4*)&w[(size_t)(n0 + row0) * K_DIM + c4];
        const float4 b1 = *(const float4*)&w[(size_t)(n0 + row1) * K_DIM + c4];
        *(float4*)&As[0][row0 * LDT + c4] = a0;
        *(float4*)&As[0][row1 * LDT + c4] = a1;
        *(float4*)&Bs[0][row0 * LDT + c4] = b0;
        *(float4*)&Bs[0][row1 * LDT + c4] = b1;
    }
    __syncthreads();

    // ---- K loop: compute buf, prefetch+store buf^1, single barrier/chunk ----
    for (int kc = 0; kc < NCH; ++kc) {
        const int buf = kc & 1;
        const bool hasNext = (kc + 1 < NCH);   // block-uniform branch

        float4 an0, an1, bn0, bn1;
        if (hasNext) {
            const int kn = (kc + 1) * BK;
            an0 = *(const float4*)&x[(size_t)(m0 + row0) * K_DIM + kn + c4];
            an1 = *(const float4*)&x[(size_t)(m0 + row1) * K_DIM + kn + c4];
            bn0 = *(const float4*)&w[(size_t)(n0 + row0) * K_DIM + kn + c4];
            bn1 = *(const float4*)&w[(size_t)(n0 + row1) * K_DIM + kn + c4];
        }

#pragma unroll
        for (int ks = 0; ks < BK; ks += 4) {
            const int ko = ks + half * 2;        // K pair: lanes<16 -> k0,k0+1
            v2f a[4], b[2];
#pragma unroll
            for (int mt = 0; mt < 4; ++mt)       // A 16x4 frag: ds_load_b64
                a[mt] = *(const v2f*)&As[buf][(wm * 64 + mt * 16 + l16) * LDT + ko];
#pragma unroll
            for (int nt = 0; nt < 2; ++nt)       // B 4x16 frag (B=W^T)
                b[nt] = *(const v2f*)&Bs[buf][(wn * 32 + nt * 16 + l16) * LDT + ko];
#pragma unroll
            for (int mt = 0; mt < 4; ++mt)
#pragma unroll
                for (int nt = 0; nt < 2; ++nt)
                    acc[mt][nt] = __builtin_amdgcn_wmma_f32_16x16x4_f32(
                        false, a[mt], false, b[nt],
                        (short)0, acc[mt][nt], false, false);
        }

        if (hasNext) {
            *(float4*)&As[buf ^ 1][row0 * LDT + c4] = an0;
            *(float4*)&As[buf ^ 1][row1 * LDT + c4] = an1;
            *(float4*)&Bs[buf ^ 1][row0 * LDT + c4] = bn0;
            *(float4*)&Bs[buf ^ 1][row1 * LDT + c4] = bn1;
        }
        __syncthreads();
    }

    // ---- epilogue: bias add + 5-step branchless lower_bound, NT stores ----
    const size_t OUT2 = (size_t)B_DIM * N_DIM;
#pragma unroll
    for (int nt = 0; nt < 2; ++nt) {
        const int lc = wn * 32 + nt * 16 + l16;  // local column in block
        const int gn = n0 + lc;
        const float bv = bias[gn];
        const float* qrow = &Qs[lc * QP];
#pragma unroll
        for (int mt = 0; mt < 4; ++mt) {
            const int gmb = m0 + wm * 64 + mt * 16 + half * 8;
#pragma unroll
            for (int r = 0; r < 8; ++r) {
                const float v = acc[mt][nt][r];
                // count of boundaries strictly < v over 32 entries (q[31]=+INF)
                int pos = (qrow[15] < v) ? 16 : 0;
                pos += (qrow[pos + 7] < v) ? 8 : 0;
                pos += (qrow[pos + 3] < v) ? 4 : 0;
                pos += (qrow[pos + 1] < v) ? 2 : 0;
                pos += (qrow[pos] < v) ? 1 : 0;
                const size_t o = (size_t)(gmb + r) * N_DIM + gn;
                __builtin_nontemporal_store(v + bv, &out[o]);
                __builtin_nontemporal_store((float)pos, &out[OUT2 + o]);
            }
        }
    }
}

extern "C" void kernel_launch(void* const* d_in, const int* in_sizes, int n_in,
                              void* d_out, int out_size, void* d_ws, size_t ws_size,
                              hipStream_t stream) {
    const float* x     = (const float*)d_in[0];
    const float* w     = (const float*)d_in[1];
    const float* bias  = (const float*)d_in[2];
    const float* quant = (const float*)d_in[3];
    float* out = (float*)d_out;
    (void)in_sizes; (void)n_in; (void)out_size; (void)d_ws; (void)ws_size;

    dim3 grid(N_DIM / BN, B_DIM / BM);   // 32 x 128 blocks, exact tiling
    eidetic_linear_kernel<<<grid, 256, 0, stream>>>(x, w, bias, quant, out);
}